// GAT_1236950581751
// MI455X (gfx1250) — compile-verified
//
#include <hip/hip_runtime.h>
#include <stdint.h>

typedef __attribute__((ext_vector_type(2))) float v2f;
typedef __attribute__((ext_vector_type(8))) float v8f;

#define NEG_SLOPE 0.2f

// ---------------------------------------------------------------------------
// float atomic-max via integer ordering trick (IEEE monotone under int/uint)
// ---------------------------------------------------------------------------
__device__ __forceinline__ void atomicMaxF(float* addr, float v) {
    if (v >= 0.0f)
        atomicMax((int*)addr, __float_as_int(v));
    else
        atomicMin((unsigned int*)addr, (unsigned int)__float_as_int(v));
}

// ---------------------------------------------------------------------------
// C[M x 64] = A[M x K] * B[K x 64] (+bias), M % 16 == 0, K % 8 == 0.
// 128 threads = 4 waves; wave w computes the 16x16 tile at columns [16w,16w+16)
// of row-tile blockIdx.x, using V_WMMA_F32_16X16X4_F32 (full f32 precision).
// A frag: lane l -> A[m = l&15][kb + {0,1}], kb = (l>>4)*2   (ISA 7.12.2)
// B frag: lane l -> B[kb + {0,1}][n = 16w + (l&15)]
// C/D:    VGPR v, lane l -> C[m = v + (l>>4)*8][n]
// ---------------------------------------------------------------------------
__global__ void gat_wmma_gemm_n64(const float* __restrict__ A,
                                  const float* __restrict__ B,
                                  float* __restrict__ C,
                                  int K,
                                  const float* __restrict__ bias) {
    const int mtile = blockIdx.x;
    const int wave  = threadIdx.x >> 5;
    const int lane  = threadIdx.x & 31;
    const int mrow  = (mtile << 4) + (lane & 15);
    const int n     = (wave << 4) + (lane & 15);
    const int kb    = (lane >> 4) << 1;            // 0 or 2

    const float* __restrict__ arow = A + (size_t)mrow * K;

    v8f acc0 = {};
    v8f acc1 = {};
    for (int k = 0; k < K; k += 8) {
        v2f a0 = *(const v2f*)(arow + k + kb);
        v2f b0;
        b0.x = B[(size_t)(k + kb) * 64 + n];
        b0.y = B[(size_t)(k + kb + 1) * 64 + n];
        acc0 = __builtin_amdgcn_wmma_f32_16x16x4_f32(false, a0, false, b0,
                                                     (short)0, acc0, false, false);
        v2f a1 = *(const v2f*)(arow + k + 4 + kb);
        v2f b1;
        b1.x = B[(size_t)(k + 4 + kb) * 64 + n];
        b1.y = B[(size_t)(k + 5 + kb) * 64 + n];
        acc1 = __builtin_amdgcn_wmma_f32_16x16x4_f32(false, a1, false, b1,
                                                     (short)0, acc1, false, false);
    }

    const float bv    = bias ? bias[n] : 0.0f;
    const int   rbase = (mtile << 4) + ((lane >> 4) << 3);
#pragma unroll
    for (int v = 0; v < 8; ++v) {
        C[(size_t)(rbase + v) * 64 + n] = acc0[v] + acc1[v] + bv;
    }
}

// ---------------------------------------------------------------------------
// Per-node attention logits for layer 1 (heads=2, C=32): thread i = n*2+h
// ---------------------------------------------------------------------------
__global__ void gat_alpha1(const float* __restrict__ h1,
                           const float* __restrict__ att_s,
                           const float* __restrict__ att_d,
                           float* __restrict__ as, float* __restrict__ ad, int N) {
    int i = blockIdx.x * blockDim.x + threadIdx.x;
    if (i >= N * 2) return;
    const int n = i >> 1, h = i & 1;
    const float* row = h1 + (size_t)n * 64 + h * 32;
    const float* ws  = att_s + h * 32;
    const float* wd  = att_d + h * 32;
    float s = 0.f, d = 0.f;
#pragma unroll
    for (int c = 0; c < 32; ++c) { s += row[c] * ws[c]; d += row[c] * wd[c]; }
    as[i] = s;
    ad[i] = d;
}

// ---------------------------------------------------------------------------
// Re-initialize all accumulators (graph replay safe)
// ---------------------------------------------------------------------------
__global__ void gat_init(float* __restrict__ out1, float* __restrict__ m1,
                         float* __restrict__ den1, float* __restrict__ out2,
                         float* __restrict__ m2, float* __restrict__ den2, int N) {
    int i = blockIdx.x * blockDim.x + threadIdx.x;
    const float ninf = -__builtin_inff();
    if (i < N * 64) out1[i] = 0.f;
    if (i < N * 3)  out2[i] = 0.f;
    if (i < N * 2)  { m1[i] = ninf; den1[i] = 0.f; }
    if (i < N)      { m2[i] = ninf; den2[i] = 0.f; }
}

__device__ __forceinline__ void edge_nodes(const long long* __restrict__ ei,
                                           long long E, long long i, int& s, int& d) {
    if (i < E) { s = (int)ei[i]; d = (int)ei[E + i]; }
    else       { s = d = (int)(i - E); }              // implicit self loops
}

// ---------------------------------------------------------------------------
// Layer-1 edge passes (heads = 2)
// ---------------------------------------------------------------------------
__global__ void gat_edge_max1(const long long* __restrict__ ei, long long E, int N,
                              const float* __restrict__ as, const float* __restrict__ ad,
                              float* __restrict__ m) {
    long long i = (long long)blockIdx.x * blockDim.x + threadIdx.x;
    if (i >= E + N) return;
    int s, d; edge_nodes(ei, E, i, s, d);
#pragma unroll
    for (int h = 0; h < 2; ++h) {
        float e = as[s * 2 + h] + ad[d * 2 + h];
        e = e > 0.f ? e : NEG_SLOPE * e;
        atomicMaxF(&m[d * 2 + h], e);
    }
}

__global__ void gat_edge_sum1(const long long* __restrict__ ei, long long E, int N,
                              const float* __restrict__ as, const float* __restrict__ ad,
                              const float* __restrict__ m, float* __restrict__ den) {
    long long i = (long long)blockIdx.x * blockDim.x + threadIdx.x;
    if (i >= E + N) return;
    int s, d; edge_nodes(ei, E, i, s, d);
#pragma unroll
    for (int h = 0; h < 2; ++h) {
        float e = as[s * 2 + h] + ad[d * 2 + h];
        e = e > 0.f ? e : NEG_SLOPE * e;
        atomicAdd(&den[d * 2 + h], expf(e - m[d * 2 + h]));
    }
}

// one thread per (edge, head); scatter-add 32 channels of h1[src]
__global__ void gat_edge_agg1(const long long* __restrict__ ei, long long E, int N,
                              const float* __restrict__ as, const float* __restrict__ ad,
                              const float* __restrict__ m, const float* __restrict__ den,
                              const float* __restrict__ h1, float* __restrict__ out1) {
    long long i = (long long)blockIdx.x * blockDim.x + threadIdx.x;
    if (i >= (E + N) * 2) return;
    const long long eidx = i >> 1;
    const int h = (int)(i & 1);
    int s, d; edge_nodes(ei, E, eidx, s, d);
    float e = as[s * 2 + h] + ad[d * 2 + h];
    e = e > 0.f ? e : NEG_SLOPE * e;
    const float w = expf(e - m[d * 2 + h]) / (den[d * 2 + h] + 1e-16f);
    const float* __restrict__ hr   = h1 + (size_t)s * 64 + h * 32;
    float* __restrict__       orow = out1 + (size_t)d * 64 + h * 32;
#pragma unroll
    for (int c = 0; c < 32; c += 4) {
        float4 v = *(const float4*)(hr + c);      // global_load_b128 gather
        atomicAdd(orow + c + 0, v.x * w);
        atomicAdd(orow + c + 1, v.y * w);
        atomicAdd(orow + c + 2, v.z * w);
        atomicAdd(orow + c + 3, v.w * w);
    }
}

// ---------------------------------------------------------------------------
// Layer 2 node work: h2 = (out1 + b1) @ W2 [64x3], plus attention logits
// ---------------------------------------------------------------------------
__global__ void gat_h2(const float* __restrict__ out1, const float* __restrict__ b1,
                       const float* __restrict__ W2,
                       const float* __restrict__ att_s2, const float* __restrict__ att_d2,
                       float* __restrict__ h2, float* __restrict__ as2,
                       float* __restrict__ ad2, int N) {
    int n = blockIdx.x * blockDim.x + threadIdx.x;
    if (n >= N) return;
    const float* __restrict__ r = out1 + (size_t)n * 64;
    float y0 = 0.f, y1 = 0.f, y2 = 0.f;
#pragma unroll 8
    for (int c = 0; c < 64; ++c) {
        const float v = r[c] + b1[c];
        y0 += v * W2[c * 3 + 0];
        y1 += v * W2[c * 3 + 1];
        y2 += v * W2[c * 3 + 2];
    }
    h2[n * 3 + 0] = y0; h2[n * 3 + 1] = y1; h2[n * 3 + 2] = y2;
    as2[n] = y0 * att_s2[0] + y1 * att_s2[1] + y2 * att_s2[2];
    ad2[n] = y0 * att_d2[0] + y1 * att_d2[1] + y2 * att_d2[2];
}

// ---------------------------------------------------------------------------
// Layer-2 edge passes (heads = 1, C = 3)
// ---------------------------------------------------------------------------
__global__ void gat_edge_max2(const long long* __restrict__ ei, long long E, int N,
                              const float* __restrict__ as, const float* __restrict__ ad,
                              float* __restrict__ m) {
    long long i = (long long)blockIdx.x * blockDim.x + threadIdx.x;
    if (i >= E + N) return;
    int s, d; edge_nodes(ei, E, i, s, d);
    float e = as[s] + ad[d];
    e = e > 0.f ? e : NEG_SLOPE * e;
    atomicMaxF(&m[d], e);
}

__global__ void gat_edge_sum2(const long long* __restrict__ ei, long long E, int N,
                              const float* __restrict__ as, const float* __restrict__ ad,
                              const float* __restrict__ m, float* __restrict__ den) {
    long long i = (long long)blockIdx.x * blockDim.x + threadIdx.x;
    if (i >= E + N) return;
    int s, d; edge_nodes(ei, E, i, s, d);
    float e = as[s] + ad[d];
    e = e > 0.f ? e : NEG_SLOPE * e;
    atomicAdd(&den[d], expf(e - m[d]));
}

__global__ void gat_edge_agg2(const long long* __restrict__ ei, long long E, int N,
                              const float* __restrict__ as, const float* __restrict__ ad,
                              const float* __restrict__ m, const float* __restrict__ den,
                              const float* __restrict__ h2, float* __restrict__ out2) {
    long long i = (long long)blockIdx.x * blockDim.x + threadIdx.x;
    if (i >= E + N) return;
    int s, d; edge_nodes(ei, E, i, s, d);
    float e = as[s] + ad[d];
    e = e > 0.f ? e : NEG_SLOPE * e;
    const float w = expf(e - m[d]) / (den[d] + 1e-16f);
    atomicAdd(&out2[d * 3 + 0], h2[s * 3 + 0] * w);
    atomicAdd(&out2[d * 3 + 1], h2[s * 3 + 1] * w);
    atomicAdd(&out2[d * 3 + 2], h2[s * 3 + 2] * w);
}

__global__ void gat_final(const float* __restrict__ out2, const float* __restrict__ b2,
                          float* __restrict__ out, int N) {
    int i = blockIdx.x * blockDim.x + threadIdx.x;
    if (i >= N * 3) return;
    out[i] = out2[i] + b2[i % 3];
}

// ---------------------------------------------------------------------------
extern "C" void kernel_launch(void* const* d_in, const int* in_sizes, int n_in,
                              void* d_out, int out_size, void* d_ws, size_t ws_size,
                              hipStream_t stream) {
    const float*     x      = (const float*)d_in[0];
    const long long* ei     = (const long long*)d_in[1];   // int64 [2, E]
    const float*     W_lin  = (const float*)d_in[2];
    const float*     b_lin  = (const float*)d_in[3];
    const float*     W1     = (const float*)d_in[4];
    const float*     att_s1 = (const float*)d_in[5];
    const float*     att_d1 = (const float*)d_in[6];
    const float*     b1     = (const float*)d_in[7];
    const float*     W2     = (const float*)d_in[8];
    const float*     att_s2 = (const float*)d_in[9];
    const float*     att_d2 = (const float*)d_in[10];
    const float*     b2     = (const float*)d_in[11];

    const int       N = in_sizes[0] / 768;       // 100000 (divisible by 16)
    const long long E = in_sizes[1] / 2;         // 3200000 raw edges
    const long long T = E + N;                   // with self loops

    // carve workspace
    float* p = (float*)d_ws;
    float* h0   = p; p += (size_t)N * 64;
    float* h1   = p; p += (size_t)N * 64;
    float* out1 = p; p += (size_t)N * 64;
    float* as1  = p; p += (size_t)N * 2;
    float* ad1  = p; p += (size_t)N * 2;
    float* m1   = p; p += (size_t)N * 2;
    float* den1 = p; p += (size_t)N * 2;
    float* h2   = p; p += (size_t)N * 3;
    float* as2  = p; p += (size_t)N;
    float* ad2  = p; p += (size_t)N;
    float* m2   = p; p += (size_t)N;
    float* den2 = p; p += (size_t)N;
    float* out2 = p; p += (size_t)N * 3;

    const int BT = 256;
    const unsigned gT  = (unsigned)((T + BT - 1) / BT);
    const unsigned gT2 = (unsigned)((T * 2 + BT - 1) / BT);

    // 1) h0 = x @ W_lin + b_lin   (WMMA f32 16x16x4, K=768)
    gat_wmma_gemm_n64<<<N / 16, 128, 0, stream>>>(x, W_lin, h0, 768, b_lin);
    // 2) h1 = h0 @ W1             (WMMA f32 16x16x4, K=64)
    gat_wmma_gemm_n64<<<N / 16, 128, 0, stream>>>(h0, W1, h1, 64, nullptr);
    // 3) per-node logits, layer 1
    gat_alpha1<<<(N * 2 + BT - 1) / BT, BT, 0, stream>>>(h1, att_s1, att_d1, as1, ad1, N);
    // 4) init accumulators
    gat_init<<<(N * 64 + BT - 1) / BT, BT, 0, stream>>>(out1, m1, den1, out2, m2, den2, N);
    // 5-7) layer-1 edge passes
    gat_edge_max1<<<gT, BT, 0, stream>>>(ei, E, N, as1, ad1, m1);
    gat_edge_sum1<<<gT, BT, 0, stream>>>(ei, E, N, as1, ad1, m1, den1);
    gat_edge_agg1<<<gT2, BT, 0, stream>>>(ei, E, N, as1, ad1, m1, den1, h1, out1);
    // 8) layer-2 node work
    gat_h2<<<(N + BT - 1) / BT, BT, 0, stream>>>(out1, b1, W2, att_s2, att_d2,
                                                 h2, as2, ad2, N);
    // 9-11) layer-2 edge passes
    gat_edge_max2<<<gT, BT, 0, stream>>>(ei, E, N, as2, ad2, m2);
    gat_edge_sum2<<<gT, BT, 0, stream>>>(ei, E, N, as2, ad2, m2, den2);
    gat_edge_agg2<<<gT, BT, 0, stream>>>(ei, E, N, as2, ad2, m2, den2, h2, out2);
    // 12) bias + output
    gat_final<<<(N * 3 + BT - 1) / BT, BT, 0, stream>>>(out2, b2, (float*)d_out, N);
}